// TopicNounAttention_72361609003622
// MI455X (gfx1250) — compile-verified
//
#include <hip/hip_runtime.h>
#include <hip/hip_bf16.h>
#include <cstdint>
#include <cstddef>

// Problem constants (match reference)
#define D_WORD  300
#define D_HID   312
#define SEQ_L   32
#define DTOPIC  128
#define THREADS 128
#define NCHUNK  (D_HID / 4)   // 78 float4 chunks per value row

typedef float v2f __attribute__((ext_vector_type(2)));
typedef float v8f __attribute__((ext_vector_type(8)));
typedef int   v4i __attribute__((ext_vector_type(4)));

// ---- CDNA5 feature gates (device pass only; host pass takes fallbacks) ----
#if defined(__AMDGCN__) && __has_builtin(__builtin_amdgcn_wmma_f32_16x16x4_f32)
#define HAS_WMMA_F32 1
#else
#define HAS_WMMA_F32 0
#endif

#if defined(__AMDGCN__) && __has_builtin(__builtin_amdgcn_global_load_async_to_lds_b128)
#define HAS_ASYNC_LDS 1
#else
#define HAS_ASYNC_LDS 0
#endif

#if HAS_ASYNC_LDS
typedef __attribute__((address_space(1))) v4i* gv4i_p;   // global (AS1)
typedef __attribute__((address_space(3))) v4i* lv4i_p;   // LDS (AS3)
#endif

// ---------------------------------------------------------------------------
// prep_kernel: single wave32.
//   q[t]  = sum_j q_w[t,j]*topic[j] + q_b[t]              (128x128 matvec)
//   w[d]  = scale * sum_t k_w[t,d]*q[t]                   (312x128 matvec)
//   bias  = scale * sum_t k_b[t]*q[t]
// Matvecs run on V_WMMA_F32_16X16X4_F32 (exact fp32 matrix path).
// f32 A 16x4 layout (ISA 7.12.2): lanes 0-15 -> M=lane, VGPR0=K0,VGPR1=K1;
// lanes 16-31 -> M=lane-16, VGPR0=K2,VGPR1=K3.
// f32 B 4x16 layout mirrored from documented f16 B: lane=N column,
// lanes 0-15 carry K={k0,k0+1}, lanes 16-31 carry K={k0+2,k0+3}.
// D column 0 -> lane 0 holds M=0..7 (c[0..7]), lane 16 holds M=8..15.
// All loads unconditional (clamped indices); lane masking via value selects
// (v_cndmask), not EXEC branches -> no saveexec in the K-loop.
// ---------------------------------------------------------------------------
__global__ __launch_bounds__(32) void prep_kernel(
    const float* __restrict__ topic, const float* __restrict__ kw,
    const float* __restrict__ kb, const float* __restrict__ qw,
    const float* __restrict__ qb, float* __restrict__ ws)
{
  __shared__ float ts[DTOPIC];   // staged topic vector
  __shared__ float q[DTOPIC];
  const int   lane  = threadIdx.x;
  const int   half  = lane >> 4;   // 0: lanes 0-15, 1: lanes 16-31
  const int   lsub  = lane & 15;
  const bool  col0  = (lsub == 0);
  const float scale = 0.08838834764831845f;  // 1/sqrt(128)

  #pragma unroll
  for (int i = lane; i < DTOPIC; i += 32) ts[i] = topic[i];
  __syncthreads();

#if HAS_WMMA_F32
  // ---- q = q_w @ topic + q_b : 8 row tiles x 32 K-steps, N=1 ----
  for (int r = 0; r < DTOPIC / 16; ++r) {
    v8f c = {0.f, 0.f, 0.f, 0.f, 0.f, 0.f, 0.f, 0.f};
    const int row = r * 16 + lsub;
    for (int k0 = 0; k0 < DTOPIC; k0 += 4) {
      const int kk = k0 + 2 * half;
      v2f a, bf;
      a.x  = qw[row * DTOPIC + kk];
      a.y  = qw[row * DTOPIC + kk + 1];
      const float t0 = ts[kk], t1 = ts[kk + 1];
      bf.x = col0 ? t0 : 0.f;
      bf.y = col0 ? t1 : 0.f;
      c = __builtin_amdgcn_wmma_f32_16x16x4_f32(false, a, false, bf,
                                                (short)0, c, false, false);
    }
    if (col0) {
      #pragma unroll
      for (int v = 0; v < 8; ++v) {
        const int rr = r * 16 + half * 8 + v;
        q[rr] = c[v] + qb[rr];
      }
    }
  }
#else
  for (int r = lane; r < DTOPIC; r += 32) {
    float acc = qb[r];
    for (int j = 0; j < DTOPIC; ++j) acc = fmaf(qw[r * DTOPIC + j], ts[j], acc);
    q[r] = acc;
  }
#endif
  __syncthreads();

#if HAS_WMMA_F32
  // ---- w = scale * k_w^T @ q : A[m][k] = kw[k*312 + m], 20 row tiles ----
  for (int r = 0; r < 20; ++r) {
    v8f c = {0.f, 0.f, 0.f, 0.f, 0.f, 0.f, 0.f, 0.f};
    const int  d0  = r * 16 + lsub;
    const bool dok = (d0 < D_HID);
    const int  dcl = dok ? d0 : 0;           // clamp: loads always in-bounds
    for (int k0 = 0; k0 < DTOPIC; k0 += 4) {
      const int kk = k0 + 2 * half;
      v2f a, bf;
      const float a0 = kw[kk * D_HID + dcl];
      const float a1 = kw[(kk + 1) * D_HID + dcl];
      a.x  = dok ? a0 : 0.f;
      a.y  = dok ? a1 : 0.f;
      const float q0 = q[kk], q1 = q[kk + 1];
      bf.x = col0 ? q0 : 0.f;
      bf.y = col0 ? q1 : 0.f;
      c = __builtin_amdgcn_wmma_f32_16x16x4_f32(false, a, false, bf,
                                                (short)0, c, false, false);
    }
    if (col0) {
      #pragma unroll
      for (int v = 0; v < 8; ++v) {
        const int d = r * 16 + half * 8 + v;
        if (d < D_HID) ws[d] = scale * c[v];
      }
    }
  }
#else
  for (int d = lane; d < D_HID; d += 32) {
    float acc = 0.f;
    for (int t = 0; t < DTOPIC; ++t) acc = fmaf(kw[t * D_HID + d], q[t], acc);
    ws[d] = scale * acc;
  }
#endif

  if (lane == 0) {
    float acc = 0.f;
    for (int t = 0; t < DTOPIC; ++t) acc = fmaf(kb[t], q[t], acc);
    ws[D_HID] = scale * acc;
  }
}

// ---------------------------------------------------------------------------
// fused_kernel: one workgroup per sample (4 waves).
//   stage value[32][312] in LDS (async global->LDS DMA) -> scores
//   -> wave32 masked softmax -> weighted output sum, all LDS reads b128.
// ---------------------------------------------------------------------------
__global__ __launch_bounds__(THREADS) void fused_kernel(
    const int* __restrict__ noun_ids, const int* __restrict__ lengths,
    const int* __restrict__ months, const int* __restrict__ enc,
    const float* __restrict__ we, const float* __restrict__ ws,
    float* __restrict__ out)
{
  __shared__ __align__(16) float val[SEQ_L][D_HID];  // 39936 B
  __shared__ __align__(16) float wv[D_HID];
  __shared__ float sc[SEQ_L];

  const int b   = blockIdx.x;
  const int tid = threadIdx.x;
  const int l   = tid >> 2;   // token 0..31 (4 threads per token)
  const int sub = tid & 3;

  // stage score vector (written by prep_kernel earlier on the same stream)
  if (tid < NCHUNK)
    *(float4*)&wv[tid * 4] = ((const float4*)ws)[tid];
  const float bias = ws[D_HID];

  const int   len  = lengths[b];
  const int   mon  = months[b];
  const float encv = (float)enc[0];

  const int     id  = noun_ids[b * SEQ_L + l];
  const float4* src = (const float4*)(we + (size_t)id * D_WORD);

  // ---- gather 32 x 300 floats into LDS (75 float4 chunks per token) ----
#if HAS_ASYNC_LDS
  for (int c = sub; c < D_WORD / 4; c += 4) {
    __builtin_amdgcn_global_load_async_to_lds_b128(
        (gv4i_p)(src + c), (lv4i_p)&val[l][c * 4], 0, 0);
  }
  if (sub == 0) {
    #pragma unroll
    for (int m = 0; m < 12; ++m) val[l][D_WORD + m] = (m == mon) ? encv : 0.f;
  }
#if __has_builtin(__builtin_amdgcn_s_wait_asynccnt)
  __builtin_amdgcn_s_wait_asynccnt(0);
#else
  asm volatile("s_wait_asynccnt 0" ::: "memory");
#endif
#else
  for (int c = sub; c < D_WORD / 4; c += 4) {
    const float4 v = src[c];
    *(float4*)&val[l][c * 4] = v;
  }
  if (sub == 0) {
    #pragma unroll
    for (int m = 0; m < 12; ++m) val[l][D_WORD + m] = (m == mon) ? encv : 0.f;
  }
#endif
  __syncthreads();

  // ---- scores[l] = value[l] . w + bias (4 threads/token, b128 LDS reads) ----
  float p = 0.f;
  for (int c = sub; c < NCHUNK; c += 4) {
    const float4 v  = *(const float4*)&val[l][c * 4];
    const float4 w4 = *(const float4*)&wv[c * 4];
    p = fmaf(v.x, w4.x, p);
    p = fmaf(v.y, w4.y, p);
    p = fmaf(v.z, w4.z, p);
    p = fmaf(v.w, w4.w, p);
  }
  p += __shfl_xor(p, 1, 4);
  p += __shfl_xor(p, 2, 4);
  if (sub == 0) sc[l] = p + bias;
  __syncthreads();

  // ---- masked softmax over L on one wave32 ----
  if (tid < SEQ_L) {
    const bool  valid = tid < len;
    const float s     = sc[tid];
    float x = valid ? s : -3.0e38f;
    #pragma unroll
    for (int off = 16; off > 0; off >>= 1) x = fmaxf(x, __shfl_xor(x, off, 32));
    float e = valid ? __expf(s - x) : 0.f;
    float t = e;
    #pragma unroll
    for (int off = 16; off > 0; off >>= 1) t += __shfl_xor(t, off, 32);
    sc[tid] = (len > 0) ? (e / t) : 0.f;   // len==0 -> output forced to 0
  }
  __syncthreads();

  // ---- out[b,:] = attn . value : one float4 chunk per thread ----
  if (tid < NCHUNK) {
    float4 acc = make_float4(0.f, 0.f, 0.f, 0.f);
    #pragma unroll
    for (int ll = 0; ll < SEQ_L; ++ll) {
      const float  a = sc[ll];
      const float4 v = *(const float4*)&val[ll][tid * 4];
      acc.x = fmaf(a, v.x, acc.x);
      acc.y = fmaf(a, v.y, acc.y);
      acc.z = fmaf(a, v.z, acc.z);
      acc.w = fmaf(a, v.w, acc.w);
    }
    ((float4*)(out + (size_t)b * D_HID))[tid] = acc;
  }
}

// ---------------------------------------------------------------------------
extern "C" void kernel_launch(void* const* d_in, const int* in_sizes, int n_in,
                              void* d_out, int out_size, void* d_ws, size_t ws_size,
                              hipStream_t stream) {
  const float* topic = (const float*)d_in[0];
  const int*   ids   = (const int*)d_in[1];
  const int*   lens  = (const int*)d_in[2];
  const int*   mons  = (const int*)d_in[3];
  const int*   enc   = (const int*)d_in[4];
  const float* we    = (const float*)d_in[5];
  const float* kw    = (const float*)d_in[6];  // [128, 312]
  const float* kb    = (const float*)d_in[7];  // [128]
  const float* qw    = (const float*)d_in[8];  // [128, 128]
  const float* qb    = (const float*)d_in[9];  // [128]
  float* out = (float*)d_out;
  float* ws  = (float*)d_ws;                   // w[312] + bias -> 1252 bytes
  const int B = in_sizes[2];                   // lengths count = batch

  prep_kernel<<<1, 32, 0, stream>>>(topic, kw, kb, qw, qb, ws);
  fused_kernel<<<B, THREADS, 0, stream>>>(ids, lens, mons, enc, we, ws, out);
  (void)n_in; (void)out_size; (void)ws_size;
}